// Nms_13125420056724
// MI455X (gfx1250) — compile-verified
//
#include <hip/hip_runtime.h>
#include <hip/hip_bf16.h>
#include <stdint.h>

// ---------------------------------------------------------------------------
// Greedy class-aware NMS for B=8, N=4096, MAX_DET=300 on MI455X (gfx1250).
//
//  * Latency/serial-chain bound (~0.8 MB total input vs 23.3 TB/s HBM), so
//    one 1024-thread workgroup (32 wave32 waves) per batch with the entire
//    per-batch state in LDS (~99 KB of the 320 KB/WGP budget).
//  * gfx1250-specific paths:
//      - TENSOR_LOAD_TO_LDS (TDM DMA, TENSORcnt) for the contiguous score
//        stage + __builtin_amdgcn_s_wait_tensorcnt.
//      - GLOBAL_LOAD_ASYNC_TO_LDS_B128 (ASYNCcnt) for the per-lane sorted
//        box gather + s_wait_asynccnt.
//      - DS atomics (ds_and_b32) for suppression-mask updates.
//      - global_prefetch_b8 for the classes row on the serial chain.
//  * WMMA is mathematically inapplicable: IoU is min/max/clamp, not A*B+C.
// ---------------------------------------------------------------------------

#define NMS_B        8
#define NMS_N        4096
#define NMS_NW       (NMS_N / 32)   /* alive-mask words */
#define NMS_MAXDET   300
#define NMS_TPB      1024
#define SCORE_THR    0.05f
#define IOU_THR      0.65f

// Output layout (all float32, concatenated flat in reference return order):
//   [0,      2400) dummy indices (-1)
//   [2400,   4800) scores
//   [4800,  14400) boxes (xyxy, class-shifted, as in reference)
//   [14400, 16800) classes
//   [16800, 16808) n_det
#define OUT_S_OFF    (NMS_B * NMS_MAXDET)             /* 2400  */
#define OUT_B_OFF    (2 * NMS_B * NMS_MAXDET)         /* 4800  */
#define OUT_C_OFF    (6 * NMS_B * NMS_MAXDET)         /* 14400 */
#define OUT_N_OFF    (7 * NMS_B * NMS_MAXDET)         /* 16800 */
#define OUT_TOTAL    (7 * NMS_B * NMS_MAXDET + NMS_B) /* 16808 */

typedef __attribute__((ext_vector_type(4))) unsigned int v4u;
typedef __attribute__((ext_vector_type(4))) int          v4i;
typedef __attribute__((ext_vector_type(8))) int          v8i;

// ---- gfx1250 async / TDM helpers ------------------------------------------
__device__ __forceinline__ unsigned lds_offset(const void* p) {
    // addrspacecast(local->flat) puts the LDS byte offset in the low 32 bits.
    return (unsigned)(unsigned long long)(uintptr_t)p;
}

__device__ __forceinline__ void async_load_b128_to_lds(unsigned lds_off,
                                                       unsigned long long gaddr) {
    asm volatile("global_load_async_to_lds_b128 %0, %1, off"
                 :: "v"(lds_off), "v"(gaddr)
                 : "memory");
}

__device__ __forceinline__ void wait_async_zero() {
    asm volatile("s_wait_asynccnt 0" ::: "memory");
}

// TDM: DMA one contiguous row of `nelem` 4-byte elements from global to LDS.
// D# layout per CDNA5 ISA (2-D tile: dim0 x 1).
__device__ __forceinline__ void tdm_load_row_f32(unsigned lds_base,
                                                 unsigned long long gaddr,
                                                 unsigned nelem) {
    v4u g0;
    g0.x = 1u;                                        // count=1 (valid), user mode
    g0.y = lds_base;                                  // lds_addr [63:32]
    g0.z = (unsigned)(gaddr & 0xFFFFFFFFull);         // global_addr[31:0]
    g0.w = (unsigned)((gaddr >> 32) & 0x1FFFFFFull)   // global_addr[56:32]
         | (2u << 30);                                // type=2 ("image")
    v8i g1;
    g1[0] = (int)(2u << 16);                          // data_size=2 -> 4 bytes
    g1[1] = (int)((nelem & 0xFFFFu) << 16);           // tensor_dim0[15:0]
    g1[2] = (int)(((nelem >> 16) & 0xFFFFu)           // tensor_dim0[31:16]
                  | (1u << 16));                      // tensor_dim1 = 1
    g1[3] = (int)((nelem & 0xFFFFu) << 16);           // tile_dim0 = nelem
    g1[4] = 1;                                        // tile_dim1 = 1
    g1[5] = (int)nelem;                               // tensor_dim0_stride low32
    g1[6] = 0;
    g1[7] = 0;
    v4i gz = {0, 0, 0, 0};                            // groups 2/3 unused (<=2D)
#if __clang_major__ >= 23
    v8i gz8 = {0, 0, 0, 0, 0, 0, 0, 0};
    __builtin_amdgcn_tensor_load_to_lds(g0, g1, gz, gz, gz8, 0);
#else
    __builtin_amdgcn_tensor_load_to_lds(g0, g1, gz, gz, 0);
#endif
    __builtin_amdgcn_s_wait_tensorcnt(0);
}

// ---------------------------------------------------------------------------
__global__ void nms_init_out_kernel(float* __restrict__ out, int n) {
    int i = blockIdx.x * blockDim.x + threadIdx.x;
    if (i < n) out[i] = (i < OUT_S_OFF) ? -1.0f : 0.0f;
}

__global__ __launch_bounds__(NMS_TPB, 1)
void nms_kernel(const float* __restrict__ scores,
                const float* __restrict__ boxes,
                const int*   __restrict__ classes,
                float*       __restrict__ out) {
    __shared__ alignas(16) float  s_key[NMS_N];     // 16 KB sort keys
    __shared__ int                s_idx[NMS_N];     // 16 KB original indices
    __shared__ alignas(16) float4 s_box[NMS_N];     // 64 KB sorted boxes
    __shared__ unsigned           s_alive[NMS_NW];  // 4096-bit alive mask
    __shared__ int                s_cur;
    __shared__ int                s_kept;

    const int b   = blockIdx.x;
    const int tid = threadIdx.x;
    const float* sc = scores  + (size_t)b * NMS_N;
    const float* bx = boxes   + (size_t)b * NMS_N * 4;
    const int*   cl = classes + (size_t)b * NMS_N;

    // Prefetch classes row (serial scalar lookups in the sweep hit cache).
    __builtin_prefetch(cl + tid * 16, 0, 1);

    // ---- Phase 1: TDM DMA of the whole score row into LDS (one wave) -----
    if (tid == 0) {
        tdm_load_row_f32(lds_offset(&s_key[0]),
                         (unsigned long long)(uintptr_t)sc, NMS_N);
    }
    __syncthreads();

    // Threshold transform + payload init.
    for (int i = tid; i < NMS_N; i += NMS_TPB) {
        float v = s_key[i];
        s_key[i] = (v > SCORE_THR) ? v : -1e30f;
        s_idx[i] = i;
    }
    __syncthreads();

    // ---- Phase 2: in-LDS bitonic sort, descending by score ---------------
    for (unsigned k = 2; k <= NMS_N; k <<= 1) {
        for (unsigned j = k >> 1; j > 0; j >>= 1) {
            for (unsigned i = tid; i < NMS_N; i += NMS_TPB) {
                unsigned ixj = i ^ j;
                if (ixj > i) {
                    float a = s_key[i], c = s_key[ixj];
                    bool desc = ((i & k) == 0);           // this run: descending
                    bool sw   = desc ? (a < c) : (a > c);
                    if (sw) {
                        s_key[i] = c; s_key[ixj] = a;
                        int t = s_idx[i]; s_idx[i] = s_idx[ixj]; s_idx[ixj] = t;
                    }
                }
            }
            __syncthreads();
        }
    }

    // ---- Phase 3: ASYNCcnt DMA-gather of sorted boxes into LDS -----------
    for (int i = tid; i < NMS_N; i += NMS_TPB) {
        int orig = s_idx[i];
        unsigned long long g = (unsigned long long)(uintptr_t)bx
                             + (unsigned long long)(unsigned)orig * 16ull;
        unsigned l = lds_offset(&s_box[0]) + (unsigned)i * 16u;
        async_load_b128_to_lds(l, g);
    }
    wait_async_zero();
    __syncthreads();

    // ---- Phase 4: alive bitmask = validity of sorted entries -------------
    for (int w = tid; w < NMS_NW; w += NMS_TPB) {
        unsigned word = 0u;
        for (int bit = 0; bit < 32; ++bit)
            word |= (s_key[(w << 5) + bit] > SCORE_THR) ? (1u << bit) : 0u;
        s_alive[w] = word;
    }
    if (tid == 0) { s_kept = 0; s_cur = -1; }
    __syncthreads();

    // ---- Phase 5: greedy sweep -------------------------------------------
    int start = 0;
    while (true) {
        if (tid == 0) {
            int found = -1;
            int wi = start >> 5;
            if (wi < NMS_NW) {
                unsigned w = s_alive[wi] & (0xFFFFFFFFu << (start & 31));
                for (;;) {
                    if (w) { found = (wi << 5) + __ffs(w) - 1; break; }
                    if (++wi >= NMS_NW) break;
                    w = s_alive[wi];
                }
            }
            s_cur = found;
        }
        __syncthreads();
        int cur = s_cur;
        if (cur < 0) break;                       // uniform exit

        float4 cb = s_box[cur];
        float  ca = (cb.z - cb.x) * (cb.w - cb.y);

        if (tid == 0) {
            int slot = s_kept;
            if (slot < NMS_MAXDET) {
                int o = b * NMS_MAXDET + slot;
                out[OUT_S_OFF + o] = s_key[cur];
                float* ob = out + OUT_B_OFF + o * 4;
                ob[0] = cb.x; ob[1] = cb.y; ob[2] = cb.z; ob[3] = cb.w;
                out[OUT_C_OFF + o] = (float)cl[s_idx[cur]];
            }
            s_kept = slot + 1;
        }

        // Suppress all later alive boxes overlapping `cur` (IoU > thr).
        for (int j = cur + 1 + tid; j < NMS_N; j += NMS_TPB) {
            unsigned wj = s_alive[j >> 5];
            if (!((wj >> (j & 31)) & 1u)) continue;
            float4 ob = s_box[j];
            float ix1 = fmaxf(cb.x, ob.x), iy1 = fmaxf(cb.y, ob.y);
            float ix2 = fminf(cb.z, ob.z), iy2 = fminf(cb.w, ob.w);
            float iw = fmaxf(ix2 - ix1, 0.0f), ih = fmaxf(iy2 - iy1, 0.0f);
            float inter = iw * ih;
            float oa = (ob.z - ob.x) * (ob.w - ob.y);
            // inter/(ca+oa-inter) > thr  <=>  inter > thr*(ca+oa-inter)
            if (inter > IOU_THR * (ca + oa - inter))
                atomicAnd(&s_alive[j >> 5], ~(1u << (j & 31)));
        }
        __syncthreads();

        // Once 300 detections are packed, nothing downstream can change any
        // output (n_det = min(total,300)), so exit the serial chain early.
        if (s_kept >= NMS_MAXDET) break;          // uniform (read post-barrier)
        start = cur + 1;
        if (start >= NMS_N) break;
    }
    __syncthreads();

    if (tid == 0) {
        int nd = s_kept < NMS_MAXDET ? s_kept : NMS_MAXDET;
        out[OUT_N_OFF + b] = (float)nd;
    }
}

extern "C" void kernel_launch(void* const* d_in, const int* in_sizes, int n_in,
                              void* d_out, int out_size, void* d_ws, size_t ws_size,
                              hipStream_t stream) {
    (void)in_sizes; (void)n_in; (void)out_size; (void)d_ws; (void)ws_size;
    const float* scores  = (const float*)d_in[0];  // [8,4096]   f32
    const float* boxes   = (const float*)d_in[1];  // [8,4096,4] f32 (class-shifted)
    const int*   classes = (const int*)d_in[2];    // [8,4096]   i32
    float*       out     = (float*)d_out;

    nms_init_out_kernel<<<(OUT_TOTAL + 255) / 256, 256, 0, stream>>>(out, OUT_TOTAL);
    nms_kernel<<<NMS_B, NMS_TPB, 0, stream>>>(scores, boxes, classes, out);
}